// DiffeomorphicTransformer_12128987644033
// MI455X (gfx1250) — compile-verified
//
#include <hip/hip_runtime.h>

// ---------------------------------------------------------------------------
// CPAB diffeomorphic transformer, fp32, for MI455X (gfx1250, wave32).
//   Kernel 1: 288 3x3 matrix exponentials via V_WMMA_F32_16X16X4_F32
//             (5 matrices block-diagonal in a 16x16 tile per wave).
//   Kernel 2: fused 50-step grid integration + bilinear interpolation,
//             Trels table staged to LDS with GLOBAL_LOAD_ASYNC_TO_LDS_B32.
// ---------------------------------------------------------------------------

#define NCX     3
#define NCY     3
#define NTRI    36
#define NVERT   25
#define NSTEPS  50
#define BATCH   8
#define CHAN    64
#define HH      256
#define WW      256
#define DIMTH   26            // 2 * 13 free vertices
#define NMAT    (BATCH * NTRI)        // 288 matrices
#define MPW     5                      // matrices (3x3 blocks) per wave tile
#define NBLK1   ((NMAT + MPW - 1) / MPW)   // 58 waves
#define KTERMS  12                     // Taylor terms (norm <= ~0.6 -> <1e-10)

typedef __attribute__((ext_vector_type(2))) float v2f;
typedef __attribute__((ext_vector_type(8))) float v8f;

// free-vertex lookup: vertex id -> theta slot (or -1 if boundary-pinned)
__device__ __constant__ int c_freeIdx[NVERT] = {
    -1, -1, -1, -1,
    -1,  0,  1, -1,
    -1,  2,  3, -1,
    -1, -1, -1, -1,
     4,  5,  6,  7,  8,  9, 10, 11, 12
};

// 1/k for the Taylor recurrence (avoids v_div sequences in the loop)
__device__ __constant__ float c_invk[KTERMS + 1] = {
    0.0f, 1.0f, 1.0f/2.0f, 1.0f/3.0f, 1.0f/4.0f, 1.0f/5.0f, 1.0f/6.0f,
    1.0f/7.0f, 1.0f/8.0f, 1.0f/9.0f, 1.0f/10.0f, 1.0f/11.0f, 1.0f/12.0f
};

__device__ __forceinline__ void vpos(int vid, float& x, float& y) {
    if (vid < 16) {                       // corner vertices, 4x4 grid
        x = (float)(vid & 3) * (1.0f / 3.0f);
        y = (float)(vid >> 2) * (1.0f / 3.0f);
    } else {                              // cell centers
        int c = vid - 16;
        x = ((float)(c % 3) + 0.5f) * (1.0f / 3.0f);
        y = ((float)(c / 3) + 0.5f) * (1.0f / 3.0f);
    }
}

__device__ __forceinline__ void trivids(int t, int& v0, int& v1, int& v2) {
    int cell = t >> 2, q = t & 3;
    int cx = cell % 3, cy = cell / 3;
    int a  = cy * 4 + cx;
    int b  = cy * 4 + cx + 1;
    int c2 = (cy + 1) * 4 + cx;
    int d  = (cy + 1) * 4 + cx + 1;
    switch (q) {
        case 0:  v0 = a;  v1 = b;  break;
        case 1:  v0 = b;  v1 = d;  break;
        case 2:  v0 = d;  v1 = c2; break;
        default: v0 = c2; v1 = a;  break;
    }
    v2 = 16 + cell;
}

// ---------------------------------------------------------------------------
// Kernel 1: one wave per block; 5 block-diagonal 3x3 matrices per 16x16 tile.
// T = expm(M) via Taylor, products done with v_wmma_f32_16x16x4_f32.
// Output: trels[m*6 + row*3 + col], m in [0,288), rows 0..1 of each expm.
// ---------------------------------------------------------------------------
__global__ void __launch_bounds__(32)
trels_expm_kernel(const float* __restrict__ theta, float* __restrict__ trels) {
    __shared__ float Mbuf[256];   // dense 16x16, row-major: M (block-diagonal)
    __shared__ float Qbuf[256];   // dense 16x16: running power M^k / k!

    const int lane = threadIdx.x;               // 0..31
    const int row  = lane & 15;
    const int hi   = lane >> 4;
    const int col  = row;                       // column for B/C/D layouts

    // zero the M tile
    #pragma unroll
    for (int v = 0; v < 8; ++v) Mbuf[lane * 8 + v] = 0.0f;
    __syncthreads();

    // scalar phase: lanes 0..4 each build one 3x3 generator M = Ahat/NSTEPS
    if (lane < MPW) {
        int m = blockIdx.x * MPW + lane;
        if (m < NMAT) {
            int b = m / NTRI, t = m % NTRI;
            int v0, v1, v2i;
            trivids(t, v0, v1, v2i);
            float X0, Y0, X1, Y1, X2, Y2;
            vpos(v0, X0, Y0); vpos(v1, X1, Y1); vpos(v2i, X2, Y2);
            // inverse of Ph = [[X0,Y0,1],[X1,Y1,1],[X2,Y2,1]]
            float a = X0, bb = Y0, d = X1, e = Y1, g = X2, h = Y2;
            float det = a * (e - h) - bb * (d - g) + (d * h - e * g);
            float rdet = 1.0f / det;
            float inv[3][3];
            inv[0][0] = (e - h) * rdet;         inv[0][1] = (h - bb) * rdet;        inv[0][2] = (bb - e) * rdet;
            inv[1][0] = (g - d) * rdet;         inv[1][1] = (a - g) * rdet;         inv[1][2] = (d - a) * rdet;
            inv[2][0] = (d * h - e * g) * rdet; inv[2][1] = (bb * g - a * h) * rdet; inv[2][2] = (a * e - bb * d) * rdet;
            // vertex velocities U[j][dim] from theta (0 for pinned vertices)
            int vids[3] = { v0, v1, v2i };
            float U[3][2];
            #pragma unroll
            for (int j = 0; j < 3; ++j) {
                int fi = c_freeIdx[vids[j]];
                U[j][0] = (fi >= 0) ? theta[b * DIMTH + fi * 2 + 0] : 0.0f;
                U[j][1] = (fi >= 0) ? theta[b * DIMTH + fi * 2 + 1] : 0.0f;
            }
            // M[dd][i2] = sum_j inv[i2][j] * U[j][dd] / NSTEPS ; row 2 stays 0
            #pragma unroll
            for (int dd = 0; dd < 2; ++dd) {
                #pragma unroll
                for (int i2 = 0; i2 < 3; ++i2) {
                    float s = inv[i2][0] * U[0][dd] + inv[i2][1] * U[1][dd]
                            + inv[i2][2] * U[2][dd];
                    Mbuf[(3 * lane + dd) * 16 + (3 * lane + i2)] = s * (1.0f / (float)NSTEPS);
                }
            }
        }
    }
    __syncthreads();

    // A fragments of M for the four K-slices (32-bit A 16x4 layout:
    // afrag[kb][v] = element(row, K = kb*4 + v + 2*hi))
    v2f afrag[4];
    #pragma unroll
    for (int kb = 0; kb < 4; ++kb) {
        afrag[kb][0] = Mbuf[row * 16 + kb * 4 + 2 * hi + 0];
        afrag[kb][1] = Mbuf[row * 16 + kb * 4 + 2 * hi + 1];
    }

    // t = I + M in C/D layout; Qbuf = M (k = 1 term)
    v8f t;
    #pragma unroll
    for (int v = 0; v < 8; ++v) {
        int r = v + 8 * hi;
        float mv = Mbuf[r * 16 + col];
        Qbuf[r * 16 + col] = mv;
        t[v] = mv + ((r == col) ? 1.0f : 0.0f);
    }
    __syncthreads();

    // Taylor: Q_k = (M x Q_{k-1}) / k, accumulate into t
    for (int k = 2; k <= KTERMS; ++k) {
        // B slices of Q: bfrag[kb][v] = element(K = kb*4 + v + 2*hi, col)
        v2f bfrag[4];
        #pragma unroll
        for (int kb = 0; kb < 4; ++kb) {
            bfrag[kb][0] = Qbuf[(kb * 4 + 0 + 2 * hi) * 16 + col];
            bfrag[kb][1] = Qbuf[(kb * 4 + 1 + 2 * hi) * 16 + col];
        }
        v8f acc = {};
        #pragma unroll
        for (int kb = 0; kb < 4; ++kb) {
            acc = __builtin_amdgcn_wmma_f32_16x16x4_f32(
                false, afrag[kb], false, bfrag[kb], (short)0, acc, false, false);
        }
        float s = c_invk[k];
        #pragma unroll
        for (int v = 0; v < 8; ++v) { acc[v] *= s; t[v] += acc[v]; }
        __syncthreads();
        #pragma unroll
        for (int v = 0; v < 8; ++v) Qbuf[(v + 8 * hi) * 16 + col] = acc[v];
        __syncthreads();
    }

    // scatter the 2x3 affine rows of each diagonal block to global memory
    #pragma unroll
    for (int v = 0; v < 8; ++v) {
        int r = v + 8 * hi;
        int mblk = col / 3;
        int j = col - 3 * mblk;
        int i2 = r - 3 * mblk;
        if (col < 15 && i2 >= 0 && i2 < 2) {
            int m = blockIdx.x * MPW + mblk;
            if (m < NMAT) trels[m * 6 + i2 * 3 + j] = t[v];
        }
    }
}

// ---------------------------------------------------------------------------
// Kernel 2: fused 50-step CPAB integration + bilinear sampling.
// One block per (batch, image row); threadIdx.x = w. The batch's 216-float
// Trels table is staged to LDS with the CDNA5 async memory->LDS path
// (GLOBAL_LOAD_ASYNC_TO_LDS_B32, tracked by ASYNCcnt), then consumed from LDS.
// NOTE: sT is the ONLY __shared__ object in this kernel, so its LDS base
// offset is 0; the per-lane LDS destination address is simply w*4.
// ---------------------------------------------------------------------------
__global__ void __launch_bounds__(256)
warp_interp_kernel(const float* __restrict__ x, const float* __restrict__ trels,
                   float* __restrict__ out) {
    const int bh   = blockIdx.x;
    const int b    = bh >> 8;      // batch
    const int hrow = bh & 255;     // image row
    const int w    = threadIdx.x;  // image col

    __shared__ float sT[NTRI * 6];           // 216 floats, LDS offset 0
    if (w < NTRI * 6) {
        unsigned lds_addr = (unsigned)(w * 4);            // byte addr in LDS
        const float* gsrc = trels + b * NTRI * 6 + w;     // per-lane global
        // copies 4 bytes from the global address gsrc to the LDS address lds_addr  (async, ASYNCcnt-tracked, no VGPR data)
        asm volatile("global_load_async_to_lds_b32 %0, %1, off"
                     :: "v"(lds_addr), "v"(gsrc) : "memory");
    }
    asm volatile("s_wait_asynccnt 0" ::: "memory");
    __syncthreads();

    float px = (float)w    * (1.0f / (float)(WW - 1));
    float py = (float)hrow * (1.0f / (float)(HH - 1));

    #pragma unroll 1
    for (int s = 0; s < NSTEPS; ++s) {
        float xc = fminf(fmaxf(px, 0.0f), 0.999999f) * 3.0f;
        float yc = fminf(fmaxf(py, 0.0f), 0.999999f) * 3.0f;
        float cxf = floorf(xc), cyf = floorf(yc);
        float xl = xc - cxf, yl = yc - cyf;
        int tri = (xl < yl) ? ((1.0f - xl < yl) ? 2 : 3)
                            : ((1.0f - xl < yl) ? 1 : 0);
        int idx = 4 * ((int)cxf + (int)cyf * 3) + tri;
        const float* T = &sT[idx * 6];
        float nx = fmaf(T[0], px, fmaf(T[1], py, T[2]));
        float ny = fmaf(T[3], px, fmaf(T[4], py, T[5]));
        px = nx; py = ny;
    }

    // bilinear sample (clamp semantics identical to the reference)
    float gx = px * (float)(WW - 1);
    float gy = py * (float)(HH - 1);
    int x0 = (int)fminf(fmaxf(floorf(gx), 0.0f), (float)(WW - 2));
    int y0 = (int)fminf(fmaxf(floorf(gy), 0.0f), (float)(HH - 2));
    float wx = fminf(fmaxf(gx - (float)x0, 0.0f), 1.0f);
    float wy = fminf(fmaxf(gy - (float)y0, 0.0f), 1.0f);
    float w00 = (1.0f - wx) * (1.0f - wy);
    float w01 = wx * (1.0f - wy);
    float w10 = (1.0f - wx) * wy;
    float w11 = wx * wy;

    const size_t plane = (size_t)HH * WW;
    const size_t base  = (size_t)b * CHAN * plane;
    const int    i00   = y0 * WW + x0;
    const size_t obase = base + (size_t)hrow * WW + (size_t)w;

    #pragma unroll 4
    for (int c = 0; c < CHAN; ++c) {
        const float* pc = x + base + (size_t)c * plane;
        float Ia = pc[i00];
        float Ib = pc[i00 + 1];
        float Ic = pc[i00 + WW];
        float Id = pc[i00 + WW + 1];
        out[obase + (size_t)c * plane] =
            fmaf(Ia, w00, fmaf(Ib, w01, fmaf(Ic, w10, Id * w11)));
    }
}

// ---------------------------------------------------------------------------
extern "C" void kernel_launch(void* const* d_in, const int* in_sizes, int n_in,
                              void* d_out, int out_size, void* d_ws, size_t ws_size,
                              hipStream_t stream) {
    const float* x      = (const float*)d_in[0];   // (8, 64, 256, 256) f32
    const float* params = (const float*)d_in[1];   // (8, 26) f32
    float* out   = (float*)d_out;                  // (8, 64, 256, 256) f32
    float* trels = (float*)d_ws;                   // 288*6 floats = 6.9 KB

    trels_expm_kernel<<<NBLK1, 32, 0, stream>>>(params, trels);
    warp_interp_kernel<<<BATCH * HH, 256, 0, stream>>>(x, trels, out);
}